// HE_ColorNormalization_65051574665513
// MI455X (gfx1250) — compile-verified
//
#include <hip/hip_runtime.h>

#define NPIX (1024 * 1024)
#define EPS 1e-8f
#define NBLK 256  // blocks in k_update / partial count

typedef float v2f __attribute__((ext_vector_type(2)));
typedef float v8f __attribute__((ext_vector_type(8)));

// ---------------- workspace layout (bytes, all 256-aligned) ----------------
#define OFF_V     0                            // 3*N floats  = 12 MB
#define OFF_WC    (3 * NPIX * 4)               // N*4 floats  = 16 MB
#define OFF_HD    (OFF_WC + NPIX * 4 * 4)      // 12 floats (padded)
#define OFF_PART  (OFF_HD + 256)               // NBLK * 28 floats
#define OFF_HIST  (OFF_PART + NBLK * 28 * 4)   // 256 u32
#define OFF_STATE (OFF_HIST + 256 * 4)         // 4 u32 (padded)
#define OFF_SCALE (OFF_STATE + 256)            // 1 float

// ---------------- V = -log(clip(pic, 0.01, 0.99)) --------------------------
__global__ void k_prep(const float* __restrict__ pic, float* __restrict__ V) {
  int base = blockIdx.x * 1024 + threadIdx.x;
#pragma unroll
  for (int k = 0; k < 4; ++k) {
    int i = base + k * 256;
    float x = pic[i];
    x = fminf(fmaxf(x, 0.01f), 0.99f);
    V[i] = -__logf(x);
  }
}

// ---------------- one NMF multiplicative-update sweep over N pixels --------
// 256 blocks x 256 threads x 16 pixels. Updates Wc in place and emits
// per-block partial sums of
//   A = V @ Wc_new (3x4)  -> acc[0..11]  (c*4+j)
//   B = Wc_new^T Wc_new   -> acc[12..27] (12 + j*4 + l)
__global__ void k_update(const float* __restrict__ V, float* __restrict__ Wc,
                         const float* __restrict__ Hd,
                         float* __restrict__ partials) {
  __shared__ float red[8][28];
  int tid = threadIdx.x;

  float hd[12];
#pragma unroll
  for (int i = 0; i < 12; ++i) hd[i] = Hd[i];

  // G = Hd^T Hd (4x4)
  float G[16];
#pragma unroll
  for (int j = 0; j < 4; ++j)
#pragma unroll
    for (int l = 0; l < 4; ++l)
      G[j * 4 + l] =
          hd[j] * hd[l] + hd[4 + j] * hd[4 + l] + hd[8 + j] * hd[8 + l];

  float acc[28];
#pragma unroll
  for (int i = 0; i < 28; ++i) acc[i] = 0.0f;

  int base = blockIdx.x * 4096 + tid;
#pragma unroll 4
  for (int k = 0; k < 16; ++k) {
    int p = base + k * 256;
    float4 w4 = ((const float4*)Wc)[p];
    float wv[4] = {w4.x, w4.y, w4.z, w4.w};
    float v0 = V[p], v1 = V[NPIX + p], v2 = V[2 * NPIX + p];

    float wn[4];
#pragma unroll
    for (int j = 0; j < 4; ++j) {
      float num = v0 * hd[j] + v1 * hd[4 + j] + v2 * hd[8 + j];
      float den = wv[0] * G[0 * 4 + j] + wv[1] * G[1 * 4 + j] +
                  wv[2] * G[2 * 4 + j] + wv[3] * G[3 * 4 + j] + EPS;
      wn[j] = wv[j] * num * __builtin_amdgcn_rcpf(den);
    }
    ((float4*)Wc)[p] = make_float4(wn[0], wn[1], wn[2], wn[3]);

#pragma unroll
    for (int j = 0; j < 4; ++j) {
      acc[j] += v0 * wn[j];
      acc[4 + j] += v1 * wn[j];
      acc[8 + j] += v2 * wn[j];
#pragma unroll
      for (int l = 0; l < 4; ++l) acc[12 + j * 4 + l] += wn[j] * wn[l];
    }
  }

  // deterministic reduction: wave32 butterfly -> LDS -> per-block partials
#pragma unroll
  for (int i = 0; i < 28; ++i) {
    float v = acc[i];
#pragma unroll
    for (int off = 16; off > 0; off >>= 1) v += __shfl_xor(v, off, 32);
    acc[i] = v;
  }
  int lane = tid & 31, wid = tid >> 5;
  if (lane == 0)
#pragma unroll
    for (int i = 0; i < 28; ++i) red[wid][i] = acc[i];
  __syncthreads();
  if (tid < 28) {
    float s = 0.0f;
#pragma unroll
    for (int w = 0; w < 8; ++w) s += red[w][tid];
    partials[blockIdx.x * 28 + tid] = s;
  }
}

// ---------------- reduce NBLK block-partials, update Hd (3x4) --------------
__global__ void k_reduce(const float* __restrict__ partials,
                         float* __restrict__ Hd) {
  __shared__ float sh[28][8];
  __shared__ float S[28];
  __shared__ float hdold[12];
  int tid = threadIdx.x;
  if (tid < 12) hdold[tid] = Hd[tid];
  int j = tid >> 3, s = tid & 7;
  if (j < 28) {
    float sum = 0.0f;
    for (int b = s; b < NBLK; b += 8) sum += partials[b * 28 + j];
    sh[j][s] = sum;
  }
  __syncthreads();
  if (tid < 28) {
    float t = 0.0f;
#pragma unroll
    for (int w = 0; w < 8; ++w) t += sh[tid][w];
    S[tid] = t;
  }
  __syncthreads();
  if (tid < 12) {
    int c = tid >> 2, jj = tid & 3;
    float a = S[c * 4 + jj];
    float den = hdold[c * 4 + 0] * S[12 + 0 * 4 + jj] +
                hdold[c * 4 + 1] * S[12 + 1 * 4 + jj] +
                hdold[c * 4 + 2] * S[12 + 2 * 4 + jj] +
                hdold[c * 4 + 3] * S[12 + 3 * 4 + jj] + EPS;
    Hd[tid] = hdold[tid] * a / den;
  }
}

// ---------------- exact 0.99-quantile via radix select ---------------------
// Concentrations are non-negative -> uint32 bit order == float order.
__global__ void k_selinit(unsigned* state, unsigned* hist, unsigned k0,
                          unsigned k1) {
  int tid = threadIdx.x;
  hist[tid] = 0u;
  if (tid == 0) {
    state[0] = 0u;  state[1] = k0;   // selection 0: prefix, remaining rank
    state[2] = 0u;  state[3] = k1;   // selection 1
  }
}

__global__ void k_selhist(const unsigned* __restrict__ data,
                          unsigned* __restrict__ hist,
                          const unsigned* __restrict__ state, int sel,
                          int shift) {
  __shared__ unsigned lh[256];
  int tid = threadIdx.x;
  lh[tid] = 0u;
  __syncthreads();
  unsigned prefix = state[sel * 2];
  unsigned mask = (shift == 24) ? 0u : (0xFFFFFFFFu << (shift + 8));
  int stride = gridDim.x * 256;
  for (int i = blockIdx.x * 256 + tid; i < 4 * NPIX; i += stride) {
    unsigned v = data[i];
    if ((v & mask) == prefix) atomicAdd(&lh[(v >> shift) & 255u], 1u);
  }
  __syncthreads();
  unsigned c = lh[tid];
  if (c) atomicAdd(&hist[tid], c);
}

__global__ void k_selscan(unsigned* hist, unsigned* state, int sel,
                          int shift) {
  int tid = threadIdx.x;
  if (tid == 0) {
    unsigned rank = state[sel * 2 + 1];
    unsigned cum = 0;
    unsigned b;
    for (b = 0; b < 256; ++b) {
      unsigned c = hist[b];
      if (cum + c > rank) { rank -= cum; break; }
      cum += c;
    }
    if (b == 256) b = 255;
    state[sel * 2] |= (b & 255u) << shift;
    state[sel * 2 + 1] = rank;
  }
  __syncthreads();
  hist[tid] = 0u;  // ready for next pass
}

__global__ void k_scale(const unsigned* __restrict__ state,
                        const float* __restrict__ htrm,
                        float* __restrict__ scalep, float frac) {
  if (threadIdx.x == 0) {
    float v0 = __uint_as_float(state[0]);
    float v1 = __uint_as_float(state[2]);
    float q = v0 + frac * (v1 - v0);
    scalep[0] = htrm[0] / q;
  }
}

// ---------------- recompose: out = clip(exp(-(Wt @ (scale*Wc^T)))) ---------
// Per 16-pixel tile: D(16x16) = A(16x4 = Wc rows) x B(4x16 = scale*Wt^T)
// via V_WMMA_F32_16X16X4_F32. A-operand load = one contiguous 256B per wave.
// Software-pipelined: 8 A loads, then 8 WMMAs, then one store section.
__global__ void k_recompose(const float* __restrict__ Wc,
                            const float* __restrict__ Wt,
                            const float* __restrict__ scalep,
                            float* __restrict__ out) {
  int tid = threadIdx.x;
  int lane = tid & 31;
  int wid = tid >> 5;
  float scale = scalep[0];

  int n = lane & 15;       // column (output channel) for this lane
  int hi = lane >> 4;      // 0: lanes 0-15 (K=0,1 / M=0-7), 1: lanes 16-31
  int k0 = hi * 2;

  // B[k][n] = scale * Wt[n][k]; v0 holds K=k0, v1 holds K=k0+1
  v2f b;
  b.x = (n < 3) ? scale * Wt[n * 4 + k0] : 0.0f;
  b.y = (n < 3) ? scale * Wt[n * 4 + k0 + 1] : 0.0f;

  int tile0 = (blockIdx.x * 8 + wid) * 8;

  v2f a[8];
#pragma unroll
  for (int t = 0; t < 8; ++t) {
    int p0 = (tile0 + t) * 16;
    // A: lane<16 -> row (p0+n) cols 0,1 ; lane>=16 -> row (p0+n) cols 2,3
    const float* rowp = Wc + (size_t)(p0 + n) * 4 + k0;
    a[t].x = rowp[0];
    a[t].y = rowp[1];
  }

  v8f c[8];
#pragma unroll
  for (int t = 0; t < 8; ++t) {
    v8f cz = {};
    c[t] = __builtin_amdgcn_wmma_f32_16x16x4_f32(false, a[t], false, b,
                                                 (short)0, cz, false, false);
  }

  if (n < 3) {
#pragma unroll
    for (int t = 0; t < 8; ++t) {
      int mbase = (tile0 + t) * 16 + hi * 8;
#pragma unroll
      for (int r = 0; r < 8; ++r) {
        float val = __expf(-c[t][r]);
        val = fminf(fmaxf(val, 0.0f), 1.0f);
        out[n * NPIX + mbase + r] = val;
      }
    }
  }
}

// ---------------------------------------------------------------------------
extern "C" void kernel_launch(void* const* d_in, const int* in_sizes, int n_in,
                              void* d_out, int out_size, void* d_ws,
                              size_t ws_size, hipStream_t stream) {
  const float* pic = (const float*)d_in[0];      // (3,1024,1024)
  const float* Wt = (const float*)d_in[1];       // (3,4)
  const float* HtRM = (const float*)d_in[2];     // scalar
  const float* W0 = (const float*)d_in[3];       // (N,4)
  const float* H0 = (const float*)d_in[4];       // (3,4)
  float* out = (float*)d_out;

  char* ws = (char*)d_ws;
  float* V = (float*)(ws + OFF_V);
  float* Wc = (float*)(ws + OFF_WC);
  float* Hd = (float*)(ws + OFF_HD);
  float* partials = (float*)(ws + OFF_PART);
  unsigned* hist = (unsigned*)(ws + OFF_HIST);
  unsigned* state = (unsigned*)(ws + OFF_STATE);
  float* scalep = (float*)(ws + OFF_SCALE);

  // working copies (inputs must not be mutated)
  hipMemcpyAsync(Wc, W0, (size_t)NPIX * 4 * sizeof(float),
                 hipMemcpyDeviceToDevice, stream);
  hipMemcpyAsync(Hd, H0, 12 * sizeof(float), hipMemcpyDeviceToDevice, stream);

  k_prep<<<3072, 256, 0, stream>>>(pic, V);

  for (int it = 0; it < 100; ++it) {
    k_update<<<NBLK, 256, 0, stream>>>(V, Wc, Hd, partials);
    k_reduce<<<1, 256, 0, stream>>>(partials, Hd);
  }

  // quantile ranks: pos = 0.99*(4N-1); need order stats at floor(pos), +1
  double pos = 0.99 * (double)(4 * NPIX - 1);
  unsigned kk = (unsigned)pos;
  float frac = (float)(pos - (double)kk);

  k_selinit<<<1, 256, 0, stream>>>(state, hist, kk, kk + 1);
  for (int sel = 0; sel < 2; ++sel) {
    for (int shift = 24; shift >= 0; shift -= 8) {
      k_selhist<<<512, 256, 0, stream>>>((const unsigned*)Wc, hist, state, sel,
                                         shift);
      k_selscan<<<1, 256, 0, stream>>>(hist, state, sel, shift);
    }
  }
  k_scale<<<1, 1, 0, stream>>>(state, HtRM, scalep, frac);

  k_recompose<<<1024, 256, 0, stream>>>(Wc, Wt, scalep, out);
}